// SGC_14929306321144
// MI455X (gfx1250) — compile-verified
//
#include <hip/hip_runtime.h>
#include <stdint.h>

typedef float v2f __attribute__((ext_vector_type(2)));
typedef float v8f __attribute__((ext_vector_type(8)));

#define DFEAT 64

// ---------- degree / norm precompute ----------

__global__ void sgc_init_deg(float* deg, int N) {
    int i = blockIdx.x * blockDim.x + threadIdx.x;
    if (i < N) deg[i] = 1.0f;  // self-loop contributes 1 to every node's in-degree
}

__global__ void sgc_deg_accum(float* deg, const long long* __restrict__ dst, int E) {
    int e = blockIdx.x * blockDim.x + threadIdx.x;
    if (e < E) unsafeAtomicAdd(&deg[(int)dst[e]], 1.0f);
}

__global__ void sgc_rsqrt_inplace(float* deg, int N) {
    int i = blockIdx.x * blockDim.x + threadIdx.x;
    if (i < N) {
        float d = deg[i];
        deg[i] = (d > 0.0f) ? rsqrtf(d) : 0.0f;
    }
}

__global__ void sgc_edge_norm(float* __restrict__ norm,
                              const long long* __restrict__ src,
                              const long long* __restrict__ dst,
                              const float* __restrict__ dinv, int E) {
    int e = blockIdx.x * blockDim.x + threadIdx.x;
    if (e < E) norm[e] = dinv[(int)src[e]] * dinv[(int)dst[e]];
}

// ---------- one propagation hop ----------
// self-loop term: xout[n] = dinv[n]^2 * xin[n]   (one float4 per thread)
__global__ void sgc_selfloop(float4* __restrict__ xout, const float4* __restrict__ xin,
                             const float* __restrict__ dinv, long total4) {
    long i = (long)blockIdx.x * blockDim.x + threadIdx.x;
    if (i >= total4) return;
    int node = (int)(i >> 4);  // 16 float4 per node (64 feats)
    float w = dinv[node];
    w = w * w;
    float4 v = xin[i];
    float4 r;
    r.x = v.x * w; r.y = v.y * w; r.z = v.z * w; r.w = v.w * w;
    xout[i] = r;
}

// edge scatter: 16 threads per edge, float4 per thread (64 features)
__global__ void sgc_scatter(float* __restrict__ xout, const float* __restrict__ xin,
                            const long long* __restrict__ src,
                            const long long* __restrict__ dst,
                            const float* __restrict__ norm, int E) {
    long t = (long)blockIdx.x * blockDim.x + threadIdx.x;
    int e = (int)(t >> 4);
    if (e >= E) return;
    int f = ((int)t & 15) * 4;
    long long s = src[e];
    long long d = dst[e];
    float w = norm[e];
    const float4 v = *(const float4*)(xin + (size_t)s * DFEAT + f);
    float* o = xout + (size_t)d * DFEAT + f;
    unsafeAtomicAdd(o + 0, w * v.x);
    unsafeAtomicAdd(o + 1, w * v.y);
    unsafeAtomicAdd(o + 2, w * v.z);
    unsafeAtomicAdd(o + 3, w * v.w);
}

// ---------- final linear via V_WMMA_F32_16X16X4_F32 ----------
// Y[m,n] = sum_k X[m,k] * W[n,k] + b[n]   (W row-major [Dout][Din])
// Block = 128 threads = 4 wave32; wave w owns col-tile n_base = 16*w,
// block owns 16 rows. A 16x4 f32 layout: lanes 0-15 hold (M=lane, K=0/1 in
// vgpr0/1), lanes 16-31 hold (M=lane-16, K=2/3). B 4x16 mirrored on N.
// C/D: vgpr v -> (M = v + 8*half, N = lane&15).
__global__ void sgc_gemm_wmma(const float* __restrict__ X, const float* __restrict__ W,
                              const float* __restrict__ bias, float* __restrict__ Y,
                              int N) {
    const int lane = threadIdx.x & 31;
    const int wave = threadIdx.x >> 5;      // 0..3 -> column tile
    const int half = lane >> 4;             // 0 or 1
    const int lp   = lane & 15;
    const int m_base = blockIdx.x * 16;
    const int n_base = wave * 16;
    if (m_base >= N) return;

    int arow = m_base + lp;
    if (arow >= N) arow = N - 1;            // clamped load; store is guarded

    const float* Aptr = X + (size_t)arow * DFEAT + 2 * half;
    const float* Bptr = W + (size_t)(n_base + lp) * DFEAT + 2 * half;

    v8f c = {};
#pragma unroll
    for (int k = 0; k < DFEAT; k += 4) {
        v2f a = *(const v2f*)(Aptr + k);
        v2f b = *(const v2f*)(Bptr + k);
        // 8 args: (neg_a, A, neg_b, B, c_mod, C, reuse_a, reuse_b)
        c = __builtin_amdgcn_wmma_f32_16x16x4_f32(false, a, false, b,
                                                  (short)0, c, false, false);
    }

    float bv = bias[n_base + lp];
#pragma unroll
    for (int v = 0; v < 8; ++v) {
        int row = m_base + v + 8 * half;
        if (row < N) Y[(size_t)row * DFEAT + n_base + lp] = c[v] + bv;
    }
}

// ---------- launch ----------

extern "C" void kernel_launch(void* const* d_in, const int* in_sizes, int n_in,
                              void* d_out, int out_size, void* d_ws, size_t ws_size,
                              hipStream_t stream) {
    const float*     x    = (const float*)d_in[0];
    const long long* ei   = (const long long*)d_in[1];   // int64 [2, E]
    const float*     W    = (const float*)d_in[2];       // [Dout, Din]
    const float*     bias = (const float*)d_in[3];       // [Dout]

    const int N = in_sizes[0] / DFEAT;
    const int E = in_sizes[1] / 2;
    const long long* src = ei;
    const long long* dst = ei + E;

    // workspace layout: buf1 | buf2 | dinv | norm
    float* buf1 = (float*)d_ws;
    float* buf2 = buf1 + (size_t)N * DFEAT;
    float* dinv = buf2 + (size_t)N * DFEAT;
    float* norm = dinv + N;
    float* out  = (float*)d_out;

    const int T = 256;

    sgc_init_deg<<<(N + T - 1) / T, T, 0, stream>>>(dinv, N);
    sgc_deg_accum<<<(E + T - 1) / T, T, 0, stream>>>(dinv, dst, E);
    sgc_rsqrt_inplace<<<(N + T - 1) / T, T, 0, stream>>>(dinv, N);
    sgc_edge_norm<<<(E + T - 1) / T, T, 0, stream>>>(norm, src, dst, dinv, E);

    const long total4 = (long)N * 16;         // float4 elements per feature matrix
    const long et     = (long)E * 16;         // 16 threads per edge

    const float* cur = x;
    float* bufs[2] = {buf1, buf2};
    for (int k = 0; k < 3; ++k) {
        float* nxt = bufs[k & 1];             // hop0->buf1, hop1->buf2, hop2->buf1
        sgc_selfloop<<<(int)((total4 + T - 1) / T), T, 0, stream>>>(
            (float4*)nxt, (const float4*)cur, dinv, total4);
        sgc_scatter<<<(int)((et + T - 1) / T), T, 0, stream>>>(
            nxt, cur, src, dst, norm, E);
        cur = nxt;
    }

    sgc_gemm_wmma<<<(N + 15) / 16, 128, 0, stream>>>(cur, W, bias, out, N);
}